// ICQT_70274254897350
// MI455X (gfx1250) — compile-verified
//
#include <hip/hip_runtime.h>
#include <math.h>

// ---------------- static problem geometry ----------------
#define N_BINS   84
#define NFRAMES  256
#define HOP      1024
#define BATCH    8
#define SR_F     44100.0
#define FMIN_F   32.7
#define TWO_PI_D 6.283185307179586476925286766559
#define MAX_ROWS 960   // KPAD upper bound (actual 880)

struct Params {
    int   wlen[N_BINS];     // per-bin window length L_k
    int   rowbase[N_BINS];  // first GEMM row of bin k (2 rows per 1024-chunk: re,im)
    float omega[N_BINS];    // 2*pi*f_k/SR
    int   ktot;             // valid rows
    int   kpad;             // rows padded to multiple of 16
    int   length;           // output samples per batch row
    int   nblocks;          // ceil(length/1024)
};

typedef float v2f __attribute__((ext_vector_type(2)));
typedef float v8f __attribute__((ext_vector_type(8)));
typedef unsigned int u32x4 __attribute__((ext_vector_type(4)));
typedef int          i32x4 __attribute__((ext_vector_type(4)));
typedef int          i32x8 __attribute__((ext_vector_type(8)));

__device__ __forceinline__ int row_to_bin(int row, const Params& p) {
    int k = 0;
    for (int i = 1; i < N_BINS; ++i) k = (row >= p.rowbase[i]) ? i : k;
    return k;
}

// ---------------- 1) basis matrix, pair-interleaved: Wp[row/2][1024][2] ----------------
// row (k,j,comp): value = comp==0 ? cos(w*t)*win(t) : -sin(w*t)*win(t), t=j*1024+r, 0 if t>=L
__global__ void fill_basis(float* __restrict__ W, Params p) {
    int r   = blockIdx.x * blockDim.x + threadIdx.x;   // 0..1023
    int row = blockIdx.y;                              // 0..kpad-1
    float val = 0.0f;
    if (row < p.ktot) {
        int k    = row_to_bin(row, p);
        int rem  = row - p.rowbase[k];
        int j    = rem >> 1;
        int comp = rem & 1;
        int t    = j * HOP + r;
        int L    = p.wlen[k];
        if (t < L) {
            float win = 0.5f - 0.5f * cosf((float)TWO_PI_D * (float)t / (float)L);
            float s, c;
            sincosf(p.omega[k] * (float)t, &s, &c);
            val = (comp == 0) ? c * win : -s * win;
        }
    }
    W[(((size_t)(row >> 1) * HOP) + r) * 2 + (row & 1)] = val;
}

// ---------------- 2) inverse norm (gather form, no atomics) ----------------
__global__ void fill_norm(float* __restrict__ ninv, Params p) {
    int pp = blockIdx.x * blockDim.x + threadIdx.x;
    if (pp >= p.length) return;
    float s = 0.0f;
    for (int k = 0; k < N_BINS; ++k) {
        int L  = p.wlen[k];
        int f1 = pp >> 10;            if (f1 > NFRAMES - 1) f1 = NFRAMES - 1;
        int f0 = (pp >= L) ? (((pp - L) >> 10) + 1) : 0;
        float sc = (float)TWO_PI_D / (float)L;
        for (int f = f0; f <= f1; ++f) {
            int t = pp - (f << 10);
            float w = 0.5f - 0.5f * cosf(sc * (float)t);
            s += w * w;
        }
    }
    ninv[pp] = (s > 1e-10f) ? (1.0f / s) : 1.0f;
}

// ---------------- 3) coefficient pack, pair-interleaved: Cp[m][row/2][16][2] ----------------
__global__ void pack_coeff(const float* __restrict__ X, float* __restrict__ C, Params p) {
    size_t tid   = (size_t)blockIdx.x * blockDim.x + threadIdx.x;
    size_t total = (size_t)p.nblocks * p.kpad * 16;
    if (tid >= total) return;
    int b   = (int)(tid & 15);
    int row = (int)((tid >> 4) % (size_t)p.kpad);
    int m   = (int)(tid / ((size_t)p.kpad * 16));
    float val = 0.0f;
    if (b < BATCH && row < p.ktot) {
        int k    = row_to_bin(row, p);
        int rem  = row - p.rowbase[k];
        int j    = rem >> 1;
        int comp = rem & 1;
        int f    = m - j;
        if (f >= 0 && f < NFRAMES)
            val = X[(((size_t)b * N_BINS + k) * NFRAMES + f) * 2 + comp];
    }
    C[(((size_t)m * (p.kpad >> 1) + (row >> 1)) * 16 + b) * 2 + (row & 1)] = val;
}

// ---------------- 4) WMMA GEMM: out[16 x 1024] = C_m[16 x K] * W[K x 1024] ----------------
// 8 waves per block m; C_m staged in LDS via TDM tensor_load_to_lds; 2 N-tiles per wave.
__global__ void icqt_wmma(const float* __restrict__ C, const float* __restrict__ W,
                          const float* __restrict__ ninv, float* __restrict__ out,
                          Params p) {
    __shared__ __align__(16) float ldsC[MAX_ROWS * 16];

    const int m    = blockIdx.x;                            // output 1024-block
    const int rt0  = (blockIdx.y * 8 + threadIdx.y) * 2;    // first of 2 adjacent 16-col tiles
    const int lane = threadIdx.x;                           // wave32 lane
    const int kpad = p.kpad;
    const float* Cm = C + (size_t)m * kpad * 16;

    // --- TDM async tensor load: stage C_m (kpad*16 floats, contiguous) into LDS ---
    if (threadIdx.y == 0) {
        unsigned           ldsa  = (unsigned)(uintptr_t)(void*)&ldsC[0];
        unsigned long long ga    = (unsigned long long)(uintptr_t)Cm;
        unsigned           nelem = (unsigned)(kpad * 16);   // 14080 elements of 4B

        u32x4 g0;
        g0.x = 1u;                                          // count=1, user mode, no gather
        g0.y = ldsa;                                        // lds_addr
        g0.z = (unsigned)(ga & 0xFFFFFFFFu);                // global_addr[31:0]
        g0.w = (unsigned)((ga >> 32) & 0x01FFFFFFu)         // global_addr[56:32]
             | (2u << 30);                                  // type=2 ("image")
        i32x8 g1;
        g1[0] = 0x00020000;                                 // wg_mask=0, data_size=4B, no flags
        g1[1] = (int)((nelem & 0xFFFFu) << 16);             // tensor_dim0[15:0]
        g1[2] = (int)(((nelem >> 16) & 0xFFFFu) | (1u << 16)); // tensor_dim0[31:16], tensor_dim1=1
        g1[3] = (int)((nelem & 0xFFFFu) << 16);             // tensor_dim1 hi=0, tile_dim0=nelem
        g1[4] = 0;                                          // tile_dim1=0, tile_dim2=0 (1-D)
        g1[5] = (int)nelem;                                 // tensor_dim0_stride[31:0]
        g1[6] = 0;                                          // stride hi, dim1_stride lo
        g1[7] = 0;
        i32x4 g2 = {0, 0, 0, 0};
        i32x4 g3 = {0, 0, 0, 0};
        i32x8 g4 = {0, 0, 0, 0, 0, 0, 0, 0};                // 6-arg toolchain: extra group, zeroed
        __builtin_amdgcn_tensor_load_to_lds(g0, g1, g2, g3, g4, 0);
        __builtin_amdgcn_s_wait_tensorcnt(0);
    }
    __syncthreads();

    const int nsel    = lane & 15;        // A: batch row M / B: column N / store column
    const int pairoff = lane >> 4;        // upper half-wave owns K+2,K+3 -> next pair-row
    const int col0 = rt0 * 16 + nsel;
    const int col1 = col0 + 16;

    const v2f* __restrict__ Ap = (const v2f*)ldsC;   // [kpad/2][16]
    const v2f* __restrict__ Wp = (const v2f*)W;      // [kpad/2][1024]

    v8f acc0 = {0.f, 0.f, 0.f, 0.f, 0.f, 0.f, 0.f, 0.f};
    v8f acc1 = {0.f, 0.f, 0.f, 0.f, 0.f, 0.f, 0.f, 0.f};

    for (int k0 = 0; k0 < kpad; k0 += 16) {
        // speculative prefetch of W pair-rows ahead (W is L2-resident)
        __builtin_prefetch((const float*)(Wp + ((size_t)((k0 >> 1) + 32) * HOP + col0)), 0, 3);
#pragma unroll
        for (int kk = 0; kk < 16; kk += 4) {
            const int pr = ((k0 + kk) >> 1) + pairoff;   // pair-row for this lane
            v2f a  = Ap[(size_t)pr * 16 + nsel];         // A[M=nsel, K=pr*2 .. +1]
            v2f b0 = Wp[(size_t)pr * HOP + col0];        // B[K, N=col0]
            v2f b1 = Wp[(size_t)pr * HOP + col1];        // B[K, N=col1]
            acc0 = __builtin_amdgcn_wmma_f32_16x16x4_f32(
                       false, a, false, b0, (short)0, acc0, false, false);
            acc1 = __builtin_amdgcn_wmma_f32_16x16x4_f32(
                       false, a, false, b1, (short)0, acc1, false, false);
        }
    }

    // D layout: lanes 0-15 hold M=v (batches 0..7), N=lane.
    if (lane < 16) {
        const int pc0 = m * HOP + col0;
        const int pc1 = m * HOP + col1;
        if (pc0 < p.length) {
            const float niv = ninv[pc0];
#pragma unroll
            for (int v = 0; v < 8; ++v)
                out[(size_t)v * p.length + pc0] = acc0[v] * niv;
        }
        if (pc1 < p.length) {
            const float niv = ninv[pc1];
#pragma unroll
            for (int v = 0; v < 8; ++v)
                out[(size_t)v * p.length + pc1] = acc1[v] * niv;
        }
    }
}

// ---------------- host launcher ----------------
extern "C" void kernel_launch(void* const* d_in, const int* in_sizes, int n_in,
                              void* d_out, int out_size, void* d_ws, size_t ws_size,
                              hipStream_t stream) {
    (void)in_sizes; (void)n_in; (void)ws_size;
    const float* X = (const float*)d_in[0];
    float* out     = (float*)d_out;

    Params p;
    const double Q = 1.0 / (pow(2.0, 1.0 / 12.0) - 1.0);
    int rows = 0;
    for (int k = 0; k < N_BINS; ++k) {
        double freq  = FMIN_F * pow(2.0, (double)k / 12.0);
        long   L     = (long)nearbyint(Q * SR_F / freq);   // round-half-even like np.round
        p.wlen[k]    = (int)L;
        p.omega[k]   = (float)(TWO_PI_D * freq / SR_F);
        p.rowbase[k] = rows;
        rows += 2 * (int)((L + HOP - 1) / HOP);
    }
    p.ktot    = rows;
    p.kpad    = (rows + 15) & ~15;
    if (p.kpad > MAX_ROWS) p.kpad = MAX_ROWS;   // static geometry: never triggers
    p.length  = out_size / BATCH;
    p.nblocks = (p.length + HOP - 1) / HOP;

    // workspace layout: W | C | ninv  (~20.5 MB total)
    float* W    = (float*)d_ws;
    float* C    = W + (size_t)p.kpad * HOP;
    float* ninv = C + (size_t)p.nblocks * p.kpad * 16;

    fill_basis<<<dim3(HOP / 256, p.kpad), 256, 0, stream>>>(W, p);
    fill_norm<<<(p.length + 255) / 256, 256, 0, stream>>>(ninv, p);
    size_t ctotal = (size_t)p.nblocks * p.kpad * 16;
    pack_coeff<<<(unsigned)((ctotal + 255) / 256), 256, 0, stream>>>(X, C, p);
    icqt_wmma<<<dim3(p.nblocks, 4), dim3(32, 8), 0, stream>>>(C, W, ninv, out, p);
}